// Discriminator_34127810134220
// MI455X (gfx1250) — compile-verified
//
#include <hip/hip_runtime.h>
#include <hip/hip_bf16.h>

// ---------------------------------------------------------------------------
// MI455X (gfx1250) fused implementation.
//
// Key optimization: (H H^T) U == H (H^T U); zeroed diagonal == subtract
// ||h_i||^2 * u_i.  The 128x128 relation matrix is never materialized.
// All per-batch GEMMs run on V_WMMA_F32_16X16X4_F32 (fp32 matrix pipe, wave32)
// with operands staged in LDS.  BatchNorm global stats force kernel splits.
// ---------------------------------------------------------------------------

#define BATCH 4096
#define NPT   128
#define FDIM  3
#define DD1   6
#define DD2   12
#define DP    16       // padded inner dim for WMMA tiles
#define BNEPS 1e-5f

typedef __attribute__((ext_vector_type(2))) float v2f;
typedef __attribute__((ext_vector_type(8))) float v8f;

struct Params {
    const float *x;
    const float *fc1_w, *fc1_b, *bn1_g, *bn1_b;
    const float *fc2_w, *fc2_b, *bn2_g, *bn2_b;
    const float *fc3_w, *fc3_b;
    const float *u_w[4], *u_b[4], *ps[4], *ph[4], *wr[4];
    const float *fc4_w, *fc4_b, *bn4_g, *bn4_b;
    const float *fc5_w, *fc5_b, *fc6_w, *fc6_b;
    const float *fc7_w, *fc7_b, *fc8_w, *fc8_b;
    float *stats;   // 6*128 floats: s1,ss1,s2,ss2,s4,ss4
    float *pre4;    // [B][N][6]
    float *out;     // [B]
};

__device__ __forceinline__ float sigf(float v) { return 1.f / (1.f + __expf(-v)); }

// ---------- front-end recompute helpers (tiny MLP, VALU) --------------------
__device__ __forceinline__ void front_pre1(const Params& p, int b, int n, float pre1[DD1]) {
    float xv[FDIM];
    const float* xp = p.x + ((size_t)b * NPT + n) * FDIM;
#pragma unroll
    for (int f = 0; f < FDIM; ++f) xv[f] = xp[f];
#pragma unroll
    for (int e = 0; e < DD1; ++e) {
        float a = p.fc1_b[e];
#pragma unroll
        for (int f = 0; f < FDIM; ++f) a += xv[f] * p.fc1_w[e * FDIM + f];
        pre1[e] = a;
    }
}

__device__ __forceinline__ void front_h1(const Params& p, int b, int n, float h1[DD1]) {
    float pre1[DD1];
    front_pre1(p, b, n, pre1);
    const float inv = 1.f / ((float)BATCH * DD1);
    float m = p.stats[n] * inv;
    float v = p.stats[128 + n] * inv - m * m;
    float is = rsqrtf(v + BNEPS);
    float g = p.bn1_g[n], bb = p.bn1_b[n];
#pragma unroll
    for (int e = 0; e < DD1; ++e) h1[e] = fmaxf((pre1[e] - m) * is * g + bb, 0.f);
}

__device__ __forceinline__ void front_pre2(const Params& p, int b, int n, float pre2[DD2]) {
    float h1[DD1];
    front_h1(p, b, n, h1);
#pragma unroll
    for (int e = 0; e < DD2; ++e) {
        float a = p.fc2_b[e];
#pragma unroll
        for (int f = 0; f < DD1; ++f) a += h1[f] * p.fc2_w[e * DD1 + f];
        pre2[e] = a;
    }
}

// ---------- kernels ---------------------------------------------------------
__global__ void k_zero(float* stats) {
    int t = threadIdx.x;
    if (t < 256) { stats[t] = 0.f; stats[t + 256] = 0.f; stats[t + 512] = 0.f; }
}

__global__ void k_stats1(Params p) {
    int n = threadIdx.x;                      // thread owns channel n
    float s = 0.f, ss = 0.f;
    for (int b = blockIdx.x; b < BATCH; b += gridDim.x) {
        float pre1[DD1];
        front_pre1(p, b, n, pre1);
#pragma unroll
        for (int e = 0; e < DD1; ++e) { s += pre1[e]; ss += pre1[e] * pre1[e]; }
    }
    atomicAdd(&p.stats[n], s);
    atomicAdd(&p.stats[128 + n], ss);
}

__global__ void k_stats2(Params p) {
    int n = threadIdx.x;
    float s = 0.f, ss = 0.f;
    for (int b = blockIdx.x; b < BATCH; b += gridDim.x) {
        float pre2[DD2];
        front_pre2(p, b, n, pre2);
#pragma unroll
        for (int e = 0; e < DD2; ++e) { s += pre2[e]; ss += pre2[e] * pre2[e]; }
    }
    atomicAdd(&p.stats[256 + n], s);
    atomicAdd(&p.stats[384 + n], ss);
}

// One workgroup (256 threads = 8 wave32) per batch element.
__global__ void k_main(Params p) {
    __shared__ float Hbuf[2][NPT * DP];   // ping-pong point features (16 KB)
    __shared__ float Ubuf[NPT * DP];      // relu(unary) (8 KB)
    __shared__ float Gpart[8 * 256];      // per-wave partial 16x16 G (8 KB)
    __shared__ float Gs[256];             // reduced G
    __shared__ float sq[NPT];             // ||h_i||^2

    const int b    = blockIdx.x;
    const int t    = threadIdx.x;
    const int lane = t & 31;
    const int wv   = t >> 5;
    const int col  = lane & 15;
    const int khalf = (lane >> 4) ? 2 : 0;   // upper half-wave holds K+2 (A/B layout)
    const int hi8   = (lane >> 4) ? 8 : 0;   // C/D layout: upper half-wave rows +8

    float* Hin  = &Hbuf[0][0];
    float* Hout = &Hbuf[1][0];

    // ---- phase A: front-end MLP -> H0 in LDS -------------------------------
    if (t < NPT) {
        const int n = t;
        float pre2[DD2];
        front_pre2(p, b, n, pre2);
        const float inv = 1.f / ((float)BATCH * DD2);
        float m2 = p.stats[256 + n] * inv;
        float v2 = p.stats[384 + n] * inv - m2 * m2;
        float is2 = rsqrtf(v2 + BNEPS);
        float g2 = p.bn2_g[n], b2 = p.bn2_b[n];
        float h2[DD2];
#pragma unroll
        for (int e = 0; e < DD2; ++e) h2[e] = fmaxf((pre2[e] - m2) * is2 * g2 + b2, 0.f);
#pragma unroll
        for (int e = 0; e < DD2; ++e) {
            float a = p.fc3_b[e];
#pragma unroll
            for (int f = 0; f < DD2; ++f) a += h2[f] * p.fc3_w[e * DD2 + f];
            Hin[n * DP + e] = sigf(a);
        }
#pragma unroll
        for (int e = DD2; e < DP; ++e) Hin[n * DP + e] = 0.f;
    }
    __syncthreads();

    // ---- 4 relation modules ------------------------------------------------
    for (int mod = 0; mod < 4; ++mod) {
        const float* uw = p.u_w[mod];
        const float* ub = p.u_b[mod];
        const float  c2 = p.wr[mod][0] * p.ps[mod][0] * p.ph[mod][0] * (1.f / (float)NPT);

        // U = relu(H @ uw^T + ub), sq_i = ||h_i||^2   (256 threads, 6 outs each)
        {
            const int n = t & 127, half = t >> 7, jb = half * DD1;
            float hn[DD2];
#pragma unroll
            for (int k = 0; k < DD2; ++k) hn[k] = Hin[n * DP + k];
#pragma unroll
            for (int jj = 0; jj < DD1; ++jj) {
                const int j = jb + jj;
                float a = ub[j];
#pragma unroll
                for (int k = 0; k < DD2; ++k) a += hn[k] * uw[j * DD2 + k];
                Ubuf[n * DP + j] = fmaxf(a, 0.f);
            }
            if (half) {
#pragma unroll
                for (int j = DD2; j < DP; ++j) Ubuf[n * DP + j] = 0.f;
            } else {
                float s = 0.f;
#pragma unroll
                for (int k = 0; k < DD2; ++k) s += hn[k] * hn[k];
                sq[n] = s;
            }
        }
        __syncthreads();

        // G_partial = H^T U over this wave's 16-point K slice (WMMA fp32)
        {
            v8f g = {};
            const int k16 = wv * 16;
#pragma unroll
            for (int kc = 0; kc < 4; ++kc) {
                const int kk = k16 + kc * 4 + khalf;
                v2f a, bm;
                a.x  = Hin[kk * DP + col];        // A[m][k] = H[k][m] (transposed read)
                a.y  = Hin[(kk + 1) * DP + col];
                bm.x = Ubuf[kk * DP + col];       // B[k][n] = U[k][n]
                bm.y = Ubuf[(kk + 1) * DP + col];
                g = __builtin_amdgcn_wmma_f32_16x16x4_f32(false, a, false, bm,
                                                          (short)0, g, false, false);
            }
#pragma unroll
            for (int r = 0; r < 8; ++r)
                Gpart[wv * 256 + (r + hi8) * 16 + col] = g[r];
        }
        __syncthreads();

        // reduce 8 partial G tiles -> Gs (one element per thread)
        {
            float s = 0.f;
#pragma unroll
            for (int w8 = 0; w8 < 8; ++w8) s += Gpart[w8 * 256 + t];
            Gs[t] = s;
        }
        __syncthreads();

        // Hnew = c2 * (H @ G - diag(sq) U), this wave's 16-row block (WMMA fp32)
        {
            v8f o = {};
            const int row = wv * 16 + col;
#pragma unroll
            for (int kc = 0; kc < 4; ++kc) {
                const int kk = kc * 4 + khalf;
                v2f a, bm;
                a.x  = Hin[row * DP + kk];        // A[m][k] = H[row][k] (row-major)
                a.y  = Hin[row * DP + kk + 1];
                bm.x = Gs[kk * 16 + col];         // B[k][n] = G[k][n]
                bm.y = Gs[(kk + 1) * 16 + col];
                o = __builtin_amdgcn_wmma_f32_16x16x4_f32(false, a, false, bm,
                                                          (short)0, o, false, false);
            }
#pragma unroll
            for (int r = 0; r < 8; ++r) {
                const int i = wv * 16 + r + hi8;
                Hout[i * DP + col] = c2 * (o[r] - sq[i] * Ubuf[i * DP + col]);
            }
        }
        __syncthreads();

        float* tmp = Hin; Hin = Hout; Hout = tmp;
    }

    // ---- fc4 -> pre4 staged to global for bn4 stats ------------------------
    if (t < NPT) {
        const int n = t;
        float hn[DD2];
#pragma unroll
        for (int k = 0; k < DD2; ++k) hn[k] = Hin[n * DP + k];
        float* o = p.pre4 + ((size_t)b * NPT + n) * DD1;
#pragma unroll
        for (int e = 0; e < DD1; ++e) {
            float a = p.fc4_b[e];
#pragma unroll
            for (int k = 0; k < DD2; ++k) a += hn[k] * p.fc4_w[e * DD2 + k];
            o[e] = a;
        }
    }
}

__global__ void k_stats4(Params p) {
    int n = threadIdx.x;
    float s = 0.f, ss = 0.f;
    for (int b = blockIdx.x; b < BATCH; b += gridDim.x) {
        const float* o = p.pre4 + ((size_t)b * NPT + n) * DD1;
#pragma unroll
        for (int e = 0; e < DD1; ++e) { float v = o[e]; s += v; ss += v * v; }
    }
    atomicAdd(&p.stats[512 + n], s);
    atomicAdd(&p.stats[640 + n], ss);
}

__global__ void k_final(Params p) {
    __shared__ float red[3 * NPT];
    const int b = blockIdx.x, n = threadIdx.x;

    const float inv = 1.f / ((float)BATCH * DD1);
    float m4 = p.stats[512 + n] * inv;
    float v4 = p.stats[640 + n] * inv - m4 * m4;
    float is4 = rsqrtf(v4 + BNEPS);
    float g4 = p.bn4_g[n], b4 = p.bn4_b[n];

    const float* o = p.pre4 + ((size_t)b * NPT + n) * DD1;
    float h[DD1];
#pragma unroll
    for (int e = 0; e < DD1; ++e) h[e] = fmaxf((o[e] - m4) * is4 * g4 + b4, 0.f);

    float q[FDIM];
#pragma unroll
    for (int c = 0; c < FDIM; ++c) {
        float a = p.fc5_b[c];
#pragma unroll
        for (int e = 0; e < DD1; ++e) a += h[e] * p.fc5_w[c * DD1 + e];
        q[c] = fmaxf(a, 0.f);
    }
    float c0 = p.fc6_b[0], g0 = p.fc7_b[0], g1 = p.fc7_b[1];
#pragma unroll
    for (int c = 0; c < FDIM; ++c) {
        c0 += q[c] * p.fc6_w[c];
        g0 += q[c] * p.fc7_w[c];
        g1 += q[c] * p.fc7_w[FDIM + c];
    }
    red[0 * NPT + n] = c0; red[1 * NPT + n] = g0; red[2 * NPT + n] = g1;

    for (int s2 = 64; s2 > 0; s2 >>= 1) {
        __syncthreads();
        if (n < s2) {
#pragma unroll
            for (int c = 0; c < 3; ++c)
                red[c * NPT + n] = fmaxf(red[c * NPT + n], red[c * NPT + n + s2]);
        }
    }
    __syncthreads();
    if (n == 0) {
        float a = p.fc8_b[0];
#pragma unroll
        for (int c = 0; c < 3; ++c) a += red[c * NPT] * p.fc8_w[c];
        p.out[b] = sigf(a);
    }
}

// ---------------------------------------------------------------------------
extern "C" void kernel_launch(void* const* d_in, const int* in_sizes, int n_in,
                              void* d_out, int out_size, void* d_ws, size_t ws_size,
                              hipStream_t stream) {
    Params p;
    p.x     = (const float*)d_in[0];
    p.fc1_w = (const float*)d_in[1];  p.fc1_b = (const float*)d_in[2];
    p.bn1_g = (const float*)d_in[3];  p.bn1_b = (const float*)d_in[4];
    p.fc2_w = (const float*)d_in[5];  p.fc2_b = (const float*)d_in[6];
    p.bn2_g = (const float*)d_in[7];  p.bn2_b = (const float*)d_in[8];
    p.fc3_w = (const float*)d_in[9];  p.fc3_b = (const float*)d_in[10];
    for (int m = 0; m < 4; ++m) {
        int base = 11 + m * 5;
        p.u_w[m] = (const float*)d_in[base + 0];
        p.u_b[m] = (const float*)d_in[base + 1];
        p.ps[m]  = (const float*)d_in[base + 2];
        p.ph[m]  = (const float*)d_in[base + 3];
        p.wr[m]  = (const float*)d_in[base + 4];
    }
    p.fc4_w = (const float*)d_in[31]; p.fc4_b = (const float*)d_in[32];
    p.bn4_g = (const float*)d_in[33]; p.bn4_b = (const float*)d_in[34];
    p.fc5_w = (const float*)d_in[35]; p.fc5_b = (const float*)d_in[36];
    p.fc6_w = (const float*)d_in[37]; p.fc6_b = (const float*)d_in[38];
    p.fc7_w = (const float*)d_in[39]; p.fc7_b = (const float*)d_in[40];
    p.fc8_w = (const float*)d_in[41]; p.fc8_b = (const float*)d_in[42];

    float* ws = (float*)d_ws;
    p.stats = ws;            // 768 floats
    p.pre4  = ws + 1024;     // B*N*6 floats (~12.6 MB)
    p.out   = (float*)d_out;

    k_zero  <<<1,     256, 0, stream>>>(p.stats);
    k_stats1<<<128,   128, 0, stream>>>(p);
    k_stats2<<<128,   128, 0, stream>>>(p);
    k_main  <<<BATCH, 256, 0, stream>>>(p);
    k_stats4<<<128,   128, 0, stream>>>(p);
    k_final <<<BATCH, 128, 0, stream>>>(p);
}